// MessagePassingGNN_1116691496963
// MI455X (gfx1250) — compile-verified
//
#include <hip/hip_runtime.h>

#define N_NODES_C 100000
#define N_EDGES_C 1600000
#define N_GRAPHS_C 512
#define CH 128
#define N_CONVS_C 4

typedef __attribute__((ext_vector_type(2))) float v2f;
typedef __attribute__((ext_vector_type(8))) float v8f;

// ---------------------------------------------------------------------------
// Dense GEMM: H[M x 128] = f(X)[M x 128] @ W[128 x 128]
//   RELU_IN  : A elements pass through relu on load
//   MEAN_IN  : A rows scaled by 1/max(aux[row],1)   (aux = pool counts)
//   BIAS     : bias[col] added to H output
//   RELU_OUT : relu on H output
//   SELFLOOP : additionally write AGG = 2*aux[row]^2 * H + bias[col] (aux = dinv)
// 256 threads = 8 waves. W staged in LDS row-pair-interleaved + swizzled so
// each B fragment is one conflict-free ds_load_b64. A fragments stream through
// a 2-deep register pipeline (chunks of 8 k-steps) -> latency hidden, no spill.
// ---------------------------------------------------------------------------
template <bool RELU_IN, bool RELU_OUT, bool BIAS, bool SELFLOOP, bool MEAN_IN>
__global__ __launch_bounds__(256) void gemm128(
    const float* __restrict__ X, const float* __restrict__ W,
    const float* __restrict__ bias, const float* __restrict__ aux,
    float* __restrict__ H, float* __restrict__ AGG, int M)
{
    // element W[2p+j][c] -> sW[p*256 + ((c + ((p&1)<<4)) & 127)*2 + j]
    __shared__ float sW[CH * CH];  // 64 KB

    const int tid = threadIdx.x;
    for (int i = tid; i < 64 * 64; i += 256) {
        int p  = i >> 6;            // row pair
        int c2 = (i & 63) << 1;     // even column
        v2f r0 = *(const v2f*)(W + (2 * p) * CH + c2);
        v2f r1 = *(const v2f*)(W + (2 * p + 1) * CH + c2);
        int cc = (c2 + ((p & 1) << 4)) & 127;
        float4 t;
        t.x = r0[0]; t.y = r1[0]; t.z = r0[1]; t.w = r1[1];
        *(float4*)(sW + p * 256 + cc * 2) = t;   // 16B aligned, conflict-free
    }
    __syncthreads();

    const int wave = tid >> 5;
    const int lane = tid & 31;
    const int half = lane >> 4;   // A: K-pair 0/1 vs 2/3 ; C/D: M rows 0-7 vs 8-15
    const int r16  = lane & 15;
    const int mtiles = M >> 4;

    for (int mt = blockIdx.x * 8 + wave; mt < mtiles; mt += gridDim.x * 8) {
        const float* xp = X + (size_t)(mt * 16 + r16) * CH + half * 2;

        float rc = 1.f;
        if (MEAN_IN) rc = 1.f / fmaxf(aux[mt * 16 + r16], 1.f);

        v8f acc[8];
#pragma unroll
        for (int n = 0; n < 8; ++n)
            acc[n] = (v8f){0.f, 0.f, 0.f, 0.f, 0.f, 0.f, 0.f, 0.f};

        // ---- 2-deep A pipeline: chunks of 8 k-steps (32 K values) ----
        v2f acur[8], anxt[8];
#pragma unroll
        for (int kk = 0; kk < 8; ++kk)
            acur[kk] = *(const v2f*)(xp + (kk << 2));

#pragma unroll
        for (int c = 0; c < 4; ++c) {
            if (c < 3) {
#pragma unroll
                for (int kk = 0; kk < 8; ++kk)
                    anxt[kk] = *(const v2f*)(xp + ((c + 1) * 32 + (kk << 2)));
            }
#pragma unroll
            for (int kk = 0; kk < 8; ++kk) {
                const int k = c * 32 + (kk << 2);
                v2f a = acur[kk];
                if (RELU_IN) { a[0] = fmaxf(a[0], 0.f); a[1] = fmaxf(a[1], 0.f); }
                if (MEAN_IN) { a[0] *= rc;              a[1] *= rc; }
                const int p   = (k >> 1) + half;       // row pair for this half
                const int swz = (p & 1) << 4;
#pragma unroll
                for (int n = 0; n < 8; ++n) {
                    const int cidx = (n * 16 + r16 + swz) & 127;
                    v2f b = *(const v2f*)(sW + p * 256 + cidx * 2);
                    acc[n] = __builtin_amdgcn_wmma_f32_16x16x4_f32(
                        false, a, false, b, (short)0, acc[n], false, false);
                }
            }
            if (c < 3) {
#pragma unroll
                for (int kk = 0; kk < 8; ++kk)
                    acur[kk] = anxt[kk];               // register rename, no copy
            }
        }

        // epilogue: write H (and optionally AGG = 2*dinv^2*H + bias)
        float slv[8];
        if (SELFLOOP) {
            const float* dp = aux + mt * 16 + half * 8;   // rows this half stores
            float4 d0 = *(const float4*)(dp);
            float4 d1 = *(const float4*)(dp + 4);
            slv[0] = 2.f * d0.x * d0.x; slv[1] = 2.f * d0.y * d0.y;
            slv[2] = 2.f * d0.z * d0.z; slv[3] = 2.f * d0.w * d0.w;
            slv[4] = 2.f * d1.x * d1.x; slv[5] = 2.f * d1.y * d1.y;
            slv[6] = 2.f * d1.z * d1.z; slv[7] = 2.f * d1.w * d1.w;
        }
        float* hp = H + (size_t)mt * 16 * CH;
        float* ap = SELFLOOP ? (AGG + (size_t)mt * 16 * CH) : nullptr;
#pragma unroll
        for (int n = 0; n < 8; ++n) {
            const int col = n * 16 + r16;
            const float bv = (BIAS || SELFLOOP) ? bias[col] : 0.f;
#pragma unroll
            for (int r = 0; r < 8; ++r) {
                const int row = r + half * 8;     // C/D: VGPR r <-> M = r + 8*half
                float v = acc[n][r];
                if (BIAS)     v += bv;
                if (RELU_OUT) v = fmaxf(v, 0.f);
                hp[row * CH + col] = v;
                if (SELFLOOP) ap[row * CH + col] = v * slv[r] + bv;
            }
        }
    }
}

// ---------------------------------------------------------------------------
// Elementwise / scatter kernels
// ---------------------------------------------------------------------------
__global__ void k_fill(float* __restrict__ p, float v, int n)
{
    int i = blockIdx.x * blockDim.x + threadIdx.x;
    if (i < n) p[i] = v;
}

__global__ void k_count_deg(const int* __restrict__ dst, float* __restrict__ deg)
{
    int e = blockIdx.x * blockDim.x + threadIdx.x;
    if (e < N_EDGES_C) atomicAdd(&deg[dst[e]], 1.f);
}

__global__ void k_rsqrt(float* __restrict__ p, int n)
{
    int i = blockIdx.x * blockDim.x + threadIdx.x;
    if (i < n) p[i] = rsqrtf(p[i]);
}

// one wave per edge: agg[dst] += h[src] * dinv[src]*dinv[dst]
__global__ __launch_bounds__(256) void k_edge(
    const float* __restrict__ H, const int* __restrict__ src,
    const int* __restrict__ dst, const float* __restrict__ dinv,
    float* __restrict__ AGG)
{
    int e = blockIdx.x * 8 + (threadIdx.x >> 5);
    if (e >= N_EDGES_C) return;
    int lane = threadIdx.x & 31;
    int s = src[e], d = dst[e];
    float w = dinv[s] * dinv[d];
    float4 h = *(const float4*)(H + (size_t)s * CH + lane * 4);
    float* ap = AGG + (size_t)d * CH + lane * 4;
    atomicAdd(ap + 0, h.x * w);
    atomicAdd(ap + 1, h.y * w);
    atomicAdd(ap + 2, h.z * w);
    atomicAdd(ap + 3, h.w * w);
}

// mean-pool accumulation (applies final ReLU on read)
__global__ __launch_bounds__(256) void k_pool(
    const float* __restrict__ X, const int* __restrict__ batch,
    float* __restrict__ sums, float* __restrict__ cnt)
{
    int node = blockIdx.x * 8 + (threadIdx.x >> 5);
    if (node >= N_NODES_C) return;
    int lane = threadIdx.x & 31;
    int g = batch[node];
    float4 v = *(const float4*)(X + (size_t)node * CH + lane * 4);
    float* sp = sums + g * CH + lane * 4;
    atomicAdd(sp + 0, fmaxf(v.x, 0.f));
    atomicAdd(sp + 1, fmaxf(v.y, 0.f));
    atomicAdd(sp + 2, fmaxf(v.z, 0.f));
    atomicAdd(sp + 3, fmaxf(v.w, 0.f));
    if (lane == 0) atomicAdd(cnt + g, 1.f);
}

// out[g] = dot(h1[g], fc2_w) + fc2_b  (wave shuffle reduction)
__global__ __launch_bounds__(256) void k_fc2(
    const float* __restrict__ H1, const float* __restrict__ w,
    const float* __restrict__ b, float* __restrict__ out)
{
    int g = blockIdx.x * 8 + (threadIdx.x >> 5);
    if (g >= N_GRAPHS_C) return;
    int lane = threadIdx.x & 31;
    float4 h  = *(const float4*)(H1 + (size_t)g * CH + lane * 4);
    float4 ww = *(const float4*)(w + lane * 4);
    float s = h.x * ww.x + h.y * ww.y + h.z * ww.z + h.w * ww.w;
#pragma unroll
    for (int off = 16; off > 0; off >>= 1) s += __shfl_down(s, off, 32);
    if (lane == 0) out[g] = s + b[0];
}

// ---------------------------------------------------------------------------
extern "C" void kernel_launch(void* const* d_in, const int* in_sizes, int n_in,
                              void* d_out, int out_size, void* d_ws, size_t ws_size,
                              hipStream_t stream)
{
    (void)in_sizes; (void)n_in; (void)out_size; (void)ws_size;

    const float* x      = (const float*)d_in[0];
    const int*   ei     = (const int*)d_in[1];
    const int*   batch  = (const int*)d_in[2];
    const float* conv_w = (const float*)d_in[4];
    const float* conv_b = (const float*)d_in[5];
    const float* fc1_w  = (const float*)d_in[6];
    const float* fc1_b  = (const float*)d_in[7];
    const float* fc2_w  = (const float*)d_in[8];
    const float* fc2_b  = (const float*)d_in[9];
    const int* srcI = ei;
    const int* dstI = ei + N_EDGES_C;

    char* ws = (char*)d_ws;
    size_t off = 0;
    auto alloc = [&](size_t bytes) -> void* {
        void* p = ws + off;
        off = (off + bytes + 255) & ~(size_t)255;
        return p;
    };
    float* dinv = (float*)alloc((size_t)N_NODES_C * 4);
    float* h    = (float*)alloc((size_t)N_NODES_C * CH * 4);
    float* bufA = (float*)alloc((size_t)N_NODES_C * CH * 4);
    float* bufB = (float*)alloc((size_t)N_NODES_C * CH * 4);
    float* sums = (float*)alloc((size_t)(N_GRAPHS_C * CH + N_GRAPHS_C) * 4);
    float* cnt  = sums + N_GRAPHS_C * CH;
    float* h1   = (float*)alloc((size_t)N_GRAPHS_C * CH * 4);
    float* out  = (float*)d_out;

    // degree -> dinv = rsqrt(deg + 2)
    k_fill<<<(N_NODES_C + 255) / 256, 256, 0, stream>>>(dinv, 2.f, N_NODES_C);
    k_count_deg<<<(N_EDGES_C + 255) / 256, 256, 0, stream>>>(dstI, dinv);
    k_rsqrt<<<(N_NODES_C + 255) / 256, 256, 0, stream>>>(dinv, N_NODES_C);

    const int gemm_blocks = (N_NODES_C / 16 + 7) / 8;   // 782
    const int node_blocks = (N_NODES_C + 7) / 8;
    const int edge_blocks = (N_EDGES_C + 7) / 8;

    // conv layer 0: GEMM writes h AND agg = 2*dinv^2*h + bias (fused self-loop)
    gemm128<false, false, false, true, false><<<gemm_blocks, 256, 0, stream>>>(
        x, conv_w, conv_b, dinv, h, bufA, N_NODES_C);
    k_edge<<<edge_blocks, 256, 0, stream>>>(h, srcI, dstI, dinv, bufA);

    // conv layers 1..3: relu fused into GEMM A-load, self-loop fused into epilogue
    float* cur = bufA;
    float* nxt = bufB;
    for (int i = 1; i < N_CONVS_C; ++i) {
        gemm128<true, false, false, true, false><<<gemm_blocks, 256, 0, stream>>>(
            cur, conv_w + (size_t)i * CH * CH, conv_b + (size_t)i * CH, dinv,
            h, nxt, N_NODES_C);
        k_edge<<<edge_blocks, 256, 0, stream>>>(h, srcI, dstI, dinv, nxt);
        float* t = cur; cur = nxt; nxt = t;
    }
    // cur = final pre-relu layer output; pool applies relu on read

    k_fill<<<(N_GRAPHS_C * CH + N_GRAPHS_C + 255) / 256, 256, 0, stream>>>(
        sums, 0.f, N_GRAPHS_C * CH + N_GRAPHS_C);
    k_pool<<<node_blocks, 256, 0, stream>>>(cur, batch, sums, cnt);

    // fc1: mean-divide fused into A-load, bias+relu fused into output
    gemm128<false, true, true, false, true><<<(N_GRAPHS_C / 16 + 7) / 8, 256, 0, stream>>>(
        sums, fc1_w, fc1_b, cnt, h1, nullptr, N_GRAPHS_C);
    k_fc2<<<(N_GRAPHS_C + 7) / 8, 256, 0, stream>>>(h1, fc2_w, fc2_b, out);
}